// NN_23742579212427
// MI455X (gfx1250) — compile-verified
//
#include <hip/hip_runtime.h>
#include <hip/hip_bf16.h>

// Fused PINN forward + 1st/2nd input derivatives via forward-mode streams.
// 7 streams (H, u0..u2, w0..w2) propagate through shared-weight GEMMs done
// with V_WMMA_F32_16X16X4_F32; pointwise tanh chain-rule fusion in registers.

typedef __attribute__((ext_vector_type(2))) float v2f;
typedef __attribute__((ext_vector_type(8))) float v8f;

#define NPTS    32768
#define HID     256
#define MT      16          // rows per block
#define LSTR    260         // padded LDS row stride (floats) -> conflict-free
#define NSTREAM 7

__device__ __forceinline__ v8f wmma4(v2f a, v2f b, v8f c) {
  // D = A(16x4,f32) * B(4x16,f32) + C(16x16,f32)
  return __builtin_amdgcn_wmma_f32_16x16x4_f32(
      /*neg_a=*/false, a, /*neg_b=*/false, b,
      /*c_mod=*/(short)0, c, /*reuse_a=*/false, /*reuse_b=*/false);
}

// Straight-line tanh: v_exp_f32 + v_rcp_f32, no branches, saturates at +-1.
__device__ __forceinline__ float fast_tanh(float x) {
  const float e = __expf(2.0f * x);                 // v_exp_f32 path
  const float r = __builtin_amdgcn_rcpf(e + 1.0f);  // v_rcp_f32
  return 1.0f - 2.0f * r;
}

__global__ void __launch_bounds__(256)
pinn_fused(const float* __restrict__ X,
           const float* __restrict__ W0, const float* __restrict__ b0,
           const float* __restrict__ W1, const float* __restrict__ b1,
           const float* __restrict__ W2, const float* __restrict__ b2,
           const float* __restrict__ W3, const float* __restrict__ b3,
           const float* __restrict__ W4, const float* __restrict__ b4,
           const float* __restrict__ lb, const float* __restrict__ ub,
           float* __restrict__ out)
{
  extern __shared__ float smem[];      // NSTREAM * MT * LSTR floats
  __shared__ float xs[MT * 3];

  const int t  = threadIdx.x;
  const int r0 = blockIdx.x * MT;

  float* S[NSTREAM];
  #pragma unroll
  for (int s = 0; s < NSTREAM; ++s) S[s] = smem + (size_t)s * MT * LSTR;

  if (t < MT * 3) xs[t] = X[(size_t)r0 * 3 + t];
  __syncthreads();

  // ---------------- layer 0: 3 -> 256 (tanh) + derivative seeding ----------
  {
    const int c = t;                              // 256 threads == 256 cols
    const float lb0 = lb[0], lb1 = lb[1], lb2 = lb[2];
    const float s0 = 2.f / (ub[0] - lb0);
    const float s1 = 2.f / (ub[1] - lb1);
    const float s2 = 2.f / (ub[2] - lb2);
    const float w0c = W0[c], w1c = W0[HID + c], w2c = W0[2 * HID + c];
    const float bc  = b0[c];
    const float zu0 = s0 * w0c, zu1 = s1 * w1c, zu2 = s2 * w2c; // du(z)/dx_k
    const float Uc  = zu0 + zu1 + zu2;
    #pragma unroll 4
    for (int r = 0; r < MT; ++r) {
      const float xn0 = s0 * (xs[r * 3 + 0] - lb0) - 1.f;
      const float xn1 = s1 * (xs[r * 3 + 1] - lb1) - 1.f;
      const float xn2 = s2 * (xs[r * 3 + 2] - lb2) - 1.f;
      const float z   = xn0 * w0c + xn1 * w1c + xn2 * w2c + bc;
      const float a   = fast_tanh(z);
      const float d1  = 1.f - a * a;        // tanh'
      const float d2U = -2.f * a * d1 * Uc; // tanh'' * U
      const int off = r * LSTR + c;
      S[0][off] = a;
      S[1][off] = d1 * zu0;  S[2][off] = d1 * zu1;  S[3][off] = d1 * zu2;
      S[4][off] = d2U * zu0; S[5][off] = d2U * zu1; S[6][off] = d2U * zu2;
    }
  }
  __syncthreads();

  // ---------------- hidden layers 1..3 : 7 GEMMs sharing W, WMMA f32 -------
  const int lane  = t & 31;
  const int wid   = t >> 5;
  const int c0    = wid * 32;          // wave's 32-column strip (2 tiles)
  const int half  = lane >> 4;         // 0 | 1
  const int l15   = lane & 15;
  const int koff  = half * 2;          // A/B K offset for this half-wave
  const int rbase = half * 8;          // C-frag row base

  const float* Wl[3] = {W1, W2, W3};
  const float* bl[3] = {b1, b2, b3};

  const float* abase[NSTREAM];
  #pragma unroll
  for (int s = 0; s < NSTREAM; ++s) abase[s] = S[s] + l15 * LSTR + koff;

  for (int l = 0; l < 3; ++l) {
    const float* __restrict__ W = Wl[l];
    const float* wb = W + (size_t)koff * HID + c0 + l15;

    v8f acc[NSTREAM][2];
    #pragma unroll
    for (int s = 0; s < NSTREAM; ++s) { acc[s][0] = 0.f; acc[s][1] = 0.f; }

    #pragma unroll 2
    for (int k = 0; k < HID; k += 4) {
      const float* wk = wb + (size_t)k * HID;
      v2f bf0, bf1;                     // B frags, shared by all 7 streams
      bf0.x = wk[0];   bf0.y = wk[HID];
      bf1.x = wk[16];  bf1.y = wk[HID + 16];
      #pragma unroll
      for (int s = 0; s < NSTREAM; ++s) {
        const v2f av = *(const v2f*)(abase[s] + k);   // ds_load_b64
        acc[s][0] = wmma4(av, bf0, acc[s][0]);
        acc[s][1] = wmma4(av, bf1, acc[s][1]);
      }
    }
    __syncthreads();   // everyone done reading prev-layer LDS

    const float* bb = bl[l];
    const float bias0 = bb[c0 + l15];
    const float bias1 = bb[c0 + 16 + l15];
    #pragma unroll
    for (int tile = 0; tile < 2; ++tile) {
      const int col = c0 + tile * 16 + l15;
      const float bv = tile ? bias1 : bias0;
      #pragma unroll
      for (int e = 0; e < 8; ++e) {
        const int off = (rbase + e) * LSTR + col;
        const float a   = fast_tanh(acc[0][tile][e] + bv);
        const float zu0 = acc[1][tile][e], zu1 = acc[2][tile][e],
                    zu2 = acc[3][tile][e];
        const float zw0 = acc[4][tile][e], zw1 = acc[5][tile][e],
                    zw2 = acc[6][tile][e];
        const float d1  = 1.f - a * a;
        const float d2U = -2.f * a * d1 * (zu0 + zu1 + zu2);
        S[0][off] = a;
        S[1][off] = d1 * zu0;
        S[2][off] = d1 * zu1;
        S[3][off] = d1 * zu2;
        S[4][off] = d1 * zw0 + d2U * zu0;
        S[5][off] = d1 * zw1 + d2U * zu1;
        S[6][off] = d1 * zw2 + d2U * zu2;
      }
    }
    __syncthreads();
  }

  // ---------------- final layer 256 -> 2 (linear) + scatter outputs --------
  {
    const int row   = t & 15;
    const int which = t >> 4;          // 0..15, 14 used
    if (which < 14) {
      int stream, i, k = 0;
      if (which < 2)      { stream = 0; i = which; }
      else if (which < 8) { const int q = which - 2; i = q / 3; k = q % 3; stream = 1 + k; }
      else                { const int q = which - 8; i = q / 3; k = q % 3; stream = 4 + k; }
      const float* src = S[stream] + row * LSTR;
      float accv = 0.f;
      #pragma unroll 8
      for (int c = 0; c < HID; ++c) accv += src[c] * W4[c * 2 + i];
      const size_t n = (size_t)(r0 + row);
      if (which < 2) {
        out[n * 2 + i] = accv + b4[i];                                   // out
      } else if (which < 8) {
        out[(size_t)2 * NPTS + (size_t)i * NPTS * 3 + n * 3 + k] = accv; // p1
      } else {
        out[(size_t)8 * NPTS + (size_t)i * NPTS * 3 + n * 3 + k] = accv; // p2
      }
    }
  }
}

extern "C" void kernel_launch(void* const* d_in, const int* in_sizes, int n_in,
                              void* d_out, int out_size, void* d_ws, size_t ws_size,
                              hipStream_t stream) {
  const float* X  = (const float*)d_in[0];
  const float* W0 = (const float*)d_in[1];
  const float* b0 = (const float*)d_in[2];
  const float* W1 = (const float*)d_in[3];
  const float* b1 = (const float*)d_in[4];
  const float* W2 = (const float*)d_in[5];
  const float* b2 = (const float*)d_in[6];
  const float* W3 = (const float*)d_in[7];
  const float* b3 = (const float*)d_in[8];
  const float* W4 = (const float*)d_in[9];
  const float* b4 = (const float*)d_in[10];
  const float* lb = (const float*)d_in[11];
  const float* ub = (const float*)d_in[12];
  float* out = (float*)d_out;

  const size_t shmem = (size_t)NSTREAM * MT * LSTR * sizeof(float); // ~114 KB
  dim3 grid(NPTS / MT), block(256);
  pinn_fused<<<grid, block, shmem, stream>>>(X, W0, b0, W1, b1, W2, b2,
                                             W3, b3, W4, b4, lb, ub, out);
}